// AssociationLayer_86981677678916
// MI455X (gfx1250) — compile-verified
//
#include <hip/hip_runtime.h>

// Problem constants (match the JAX reference)
#define TP 257              // T_MAX + 1
#define DP 257              // D_MAX + 1
#define KS 257              // K row stride in floats (odd -> LDS bank rotation)
#define LPAD (TP * DP)      // 66049 flattened outputs per example
#define VPAD 260            // padded vector length (multiple of 4 for k-chunks)
#define NITER 100
#define LAMB 10.0f
#define EPSV 1e-12f
#define NTHREADS 512        // 16 waves (wave32)

// LDS layout (all offsets in floats). Total sized so every strided WMMA tile
// read (max index 69906) stays inside the allocation -> no UB, and since we
// zero the whole block first, any "overhang" read is a finite value that is
// multiplied by a zero pad.
#define KOFF   0
#define UOFF   (TP * KS)           // 66049
#define VOFF   (UOFF + VPAD)       // 66309
#define TMPOFF (VOFF + VPAD)       // 66569
#define RMOFF  (TMPOFF + VPAD)     // 66829
#define CMOFF  (RMOFF + TP)        // 67086
#define RHOFF  (CMOFF + TP)        // 67343
#define CHOFF  (RHOFF + TP)        // 67600
#define SMTOT  69920               // 279680 bytes < 320KB WGP LDS (compile-verified)

typedef float v2f __attribute__((ext_vector_type(2)));
typedef float v8f __attribute__((ext_vector_type(8)));

#define HAVE_WMMA_F32 __has_builtin(__builtin_amdgcn_wmma_f32_16x16x4_f32)

#if __has_builtin(__builtin_amdgcn_global_load_async_to_lds_b32)
#define HAVE_ASYNC_LDS 1
#else
#define HAVE_ASYNC_LDS 0
#endif

// Builtin signature (from hipcc diagnostic): first param is
// '__device__ int *' == int __attribute__((address_space(1)))*, i.e. the
// async-load builtins take int pointers in explicit address spaces.
typedef __attribute__((address_space(1))) int gas_int;  // global
typedef __attribute__((address_space(3))) int las_int;  // LDS

__device__ __forceinline__ void wait_async_zero() {
#if __has_builtin(__builtin_amdgcn_s_wait_asynccnt)
  __builtin_amdgcn_s_wait_asynccnt(0);
#else
  asm volatile("s_wait_asynccnt 0x0" ::: "memory");
#endif
}

// transport value: identical expression tree everywhere so '==' tests between
// passes are bit-exact (pure multiplies -> no FMA contraction ambiguity)
#define TVAL(r, c) ((u_[(r)] * Kb[(r) * KS + (c)]) * v_[(c)])

__global__ __launch_bounds__(NTHREADS)
void sinkhorn_assoc_gfx1250(const float* __restrict__ aff,
                            const int* __restrict__ ndet,
                            const int* __restrict__ ntrk,
                            float* __restrict__ out, int nB) {
  __shared__ float smem[SMTOT];
  float* Kb   = smem + KOFF;
  float* u_   = smem + UOFF;
  float* v_   = smem + VOFF;
  float* tm   = smem + TMPOFF;
  float* rmax = smem + RMOFF;
  float* cmax = smem + CMOFF;
  float* rhas = smem + RHOFF;
  float* chas = smem + CHOFF;

  const int b    = blockIdx.x;
  const int tid  = threadIdx.x;
  const int lane = tid & 31;
  const int wave = tid >> 5;
  const int nd   = ndet[b];
  const int nt   = ntrk[b];
  const float* A = aff + (size_t)b * (256 * 256);

  // Prefetch the affinity block toward L2 while we zero LDS.
  for (int off = tid * 128; off < 256 * 256 * 4; off += NTHREADS * 128)
    __builtin_prefetch((const char*)A + off, 0, 1);

  // Zero entire LDS block (keeps pad regions finite/zero forever).
  for (int i = tid; i < SMTOT; i += NTHREADS) smem[i] = 0.0f;
  __syncthreads();

#if HAVE_ASYNC_LDS
  // --- CDNA5 async global->LDS DMA of raw affinity, stride-257 layout ---
  // B32 per element: LDS row stride 1028B is only 4B-aligned, and B32 keeps
  // the perfect (r+c)%64 bank rotation for the 200 matvec passes.
  for (int idx = tid; idx < 256 * 256; idx += NTHREADS) {
    int r = idx >> 8, c = idx & 255;
    __builtin_amdgcn_global_load_async_to_lds_b32(
        (gas_int*)(A + idx), (las_int*)(Kb + r * KS + c), 0, 0);
  }
  wait_async_zero();   // my LDS writes have landed
  __syncthreads();     // everyone's have
  // In-place transform: raw affinity -> K kernel value.
  for (int idx = tid; idx < 256 * 256; idx += NTHREADS) {
    int r = idx >> 8, c = idx & 255;
    float raw = Kb[r * KS + c];
    float kv;
    if (r < nt && c < nd) kv = expf(LAMB * raw);
    else                  kv = (r <= nt && c <= nd) ? 1.0f : 0.0f;
    Kb[r * KS + c] = kv;
  }
#else
  // Fallback: direct VGPR path. interior = exp(lambda*aff); valid border = 1.
  for (int idx = tid; idx < 256 * 256; idx += NTHREADS) {
    int r = idx >> 8, c = idx & 255;
    float kv;
    if (r < nt && c < nd) kv = expf(LAMB * A[idx]);
    else                  kv = (r <= nt && c <= nd) ? 1.0f : 0.0f;
    Kb[r * KS + c] = kv;
  }
#endif
  for (int r = tid; r < TP; r += NTHREADS)          // border column c = 256
    Kb[r * KS + 256] = (r <= nt && nd == 256) ? 1.0f : 0.0f;
  for (int c = tid; c < 256; c += NTHREADS)         // border row r = 256
    Kb[256 * KS + c] = (nt == 256 && c <= nd) ? 1.0f : 0.0f;
  for (int c = tid; c < TP; c += NTHREADS)          // v0 = col_valid
    v_[c] = (c <= nd) ? 1.0f : 0.0f;
  __syncthreads();

  const int ml = lane & 15;          // row (A) / col (B) within tile
  const int kh = (lane >> 4) << 1;   // k half: 0 or 2

  for (int it = 0; it < NITER; ++it) {
    // ---- phase 1: tm = K @ v ------------------------------------------
#if HAVE_WMMA_F32
    for (int s = wave; s < 17; s += NTHREADS / 32) {
      const int rbase = s * 16;
      const float* krow = Kb + (rbase + ml) * KS + kh;
      v8f acc = {0.f, 0.f, 0.f, 0.f, 0.f, 0.f, 0.f, 0.f};
      for (int k0 = 0; k0 < VPAD; k0 += 4) {
        v2f a, bv;
        a.x  = krow[k0];                 // A[m][kh+0]
        a.y  = krow[k0 + 1];             // A[m][kh+1]
        bv.x = v_[k0 + kh];              // B[kh+0][*] = v (broadcast cols)
        bv.y = v_[k0 + kh + 1];
        acc = __builtin_amdgcn_wmma_f32_16x16x4_f32(
            false, a, false, bv, (short)0, acc, false, false);
      }
      if (ml == 0) {                     // lanes 0 and 16 hold column N=0
        int rb = rbase + (lane >> 4) * 8;
        #pragma unroll
        for (int i = 0; i < 8; ++i)
          if (rb + i < TP) tm[rb + i] = acc[i];
      }
    }
#else
    for (int r = tid; r < TP; r += NTHREADS) {
      float s = 0.0f;
      for (int c = 0; c < TP; ++c) s = fmaf(Kb[r * KS + c], v_[c], s);
      tm[r] = s;
    }
#endif
    __syncthreads();
    for (int r = tid; r < TP; r += NTHREADS) {
      float rs = (r < nt) ? 1.0f : ((r == nt) ? (float)nd : 0.0f);
      u_[r] = (r <= nt) ? rs / (tm[r] + EPSV) : 0.0f;
    }
    __syncthreads();

    // ---- phase 2: tm = K^T @ u ----------------------------------------
#if HAVE_WMMA_F32
    for (int s = wave; s < 17; s += NTHREADS / 32) {
      const int cbase = s * 16;
      v8f acc = {0.f, 0.f, 0.f, 0.f, 0.f, 0.f, 0.f, 0.f};
      for (int k0 = 0; k0 < VPAD; k0 += 4) {
        v2f a, bv;
        a.x  = u_[k0 + kh];                           // A = u broadcast rows
        a.y  = u_[k0 + kh + 1];
        bv.x = Kb[(k0 + kh) * KS + cbase + ml];       // B[k][n] = K[k][cbase+n]
        bv.y = Kb[(k0 + kh + 1) * KS + cbase + ml];
        acc = __builtin_amdgcn_wmma_f32_16x16x4_f32(
            false, a, false, bv, (short)0, acc, false, false);
      }
      if (lane < 16 && cbase + lane < DP) tm[cbase + lane] = acc[0];
    }
#else
    for (int c = tid; c < DP; c += NTHREADS) {
      float s = 0.0f;
      for (int r = 0; r < TP; ++r) s = fmaf(Kb[r * KS + c], u_[r], s);
      tm[c] = s;
    }
#endif
    __syncthreads();
    for (int c = tid; c < DP; c += NTHREADS) {
      float cs = (c < nd) ? 1.0f : ((c == nd) ? (float)nt : 0.0f);
      v_[c] = (c <= nd) ? cs / (tm[c] + EPSV) : 0.0f;
    }
    __syncthreads();
  }

  // ---- mutual-argmax assignment --------------------------------------
  for (int r = tid; r < TP; r += NTHREADS) {
    float m = -INFINITY;
    if (r < nt)
      for (int c = 0; c < nd; ++c) m = fmaxf(m, TVAL(r, c));
    rmax[r] = m;
  }
  for (int c = tid; c < DP; c += NTHREADS) {
    float m = -INFINITY;
    if (c < nd)
      for (int r = 0; r < nt; ++r) m = fmaxf(m, TVAL(r, c));
    cmax[c] = m;
  }
  __syncthreads();
  for (int r = tid; r < TP; r += NTHREADS) {
    float has = 0.0f;
    if (r < nt) {
      float rm = rmax[r];
      for (int c = 0; c < nd; ++c) {
        float t = TVAL(r, c);
        if (t == rm && t == cmax[c]) has = 1.0f;
      }
    }
    rhas[r] = has;
  }
  for (int c = tid; c < DP; c += NTHREADS) {
    float has = 0.0f;
    if (c < nd) {
      float cm = cmax[c];
      for (int r = 0; r < nt; ++r) {
        float t = TVAL(r, c);
        if (t == rmax[r] && t == cm) has = 1.0f;
      }
    }
    chas[c] = has;
  }
  __syncthreads();

  // ---- flatten (nt+1)x(nd+1) submatrix, zero-pad to LPAD -------------
  const int ndp = nd + 1;
  const int length = (nt + 1) * ndp;
  float* outT = out + (size_t)b * LPAD;
  float* outA = out + ((size_t)nB + (size_t)b) * LPAD;
  for (int k = tid; k < LPAD; k += NTHREADS) {
    float tv = 0.0f, av = 0.0f;
    if (k < length) {
      int r = k / ndp;
      int c = k - r * ndp;
      tv = TVAL(r, c);
      bool a;
      if (r < nt && c < nd)      a = (tv == rmax[r]) && (tv == cmax[c]);
      else if (r < nt)           a = (rhas[r] == 0.0f);   // death col c==nd
      else if (c < nd)           a = (chas[c] == 0.0f);   // birth row r==nt
      else                       a = false;
      av = a ? 1.0f : 0.0f;
    }
    outT[k] = tv;
    outA[k] = av;
  }
}

extern "C" void kernel_launch(void* const* d_in, const int* in_sizes, int n_in,
                              void* d_out, int out_size, void* d_ws, size_t ws_size,
                              hipStream_t stream) {
  const float* aff  = (const float*)d_in[0];
  const int*   ndet = (const int*)d_in[1];
  const int*   ntrk = (const int*)d_in[2];
  float*       out  = (float*)d_out;
  const int nB = in_sizes[1];   // batch size (one workgroup per example)

  sinkhorn_assoc_gfx1250<<<nB, NTHREADS, 0, stream>>>(aff, ndet, ntrk, out, nB);
}